// non_local_smooth_loss_59674275610696
// MI455X (gfx1250) — compile-verified
//
#include <hip/hip_runtime.h>
#include <cstdint>

// ---------------------------------------------------------------------------
// Non-local smoothness loss on MI455X (gfx1250, wave32).
// Box filter (window 2R+1, R=51, truncated at borders) implemented as a
// band-limited fp32 WMMA GEMM against a generated banded 0/1 mask matrix:
//   Out = M * In * M,  M[i,j] = (|i-j| <= R)
// The mask operand is synthesized in registers (zero memory traffic).
// Raw input strips are staged into LDS by the Tensor Data Mover (TDM), then
// transformed into a second, odd-stride LDS buffer so WMMA A-fragment reads
// are adjacent (ds_load_2addr) and bank-conflict-free.
// ---------------------------------------------------------------------------

#define R_WIN   51
#define IMG     512            // padded pitch and row count per channel image
#define IMG_S   (512 * 512)    // per-image stride in floats
#define HV      510            // valid H == valid W
#define NIMG    96             // 8 batches * 12 channels
#define NBINS   10             // B in the reference
#define RAWP    516            // TDM LDS row stride (512 data + 2x2 pad dwords)
#define FP      521            // transformed-strip row stride (odd)

typedef float v2f __attribute__((ext_vector_type(2)));
typedef float v8f __attribute__((ext_vector_type(8)));
typedef unsigned int v4u __attribute__((ext_vector_type(4)));
typedef int v4i __attribute__((ext_vector_type(4)));
typedef int v8i __attribute__((ext_vector_type(8)));

static __device__ __forceinline__ v8f wmma_f32x4(v2f a, v2f b, v8f c) {
  // D = A(16x4) * B(4x16) + C(16x16), fp32, wave32.
  return __builtin_amdgcn_wmma_f32_16x16x4_f32(false, a, false, b,
                                               (short)0, c, false, false);
}

static __device__ __forceinline__ int iabs(int x) { return x < 0 ? -x : x; }

// pow(x, -1.2) for x > 0 via native transcendentals
static __device__ __forceinline__ float powm12(float x) {
  return __expf(-1.2f * __logf(x));
}

// ---------------------------------------------------------------------------
// K1: dd[n, q*3+c, h, w] second-order differences of pred_map.
//  q=0: dx2[:,1:-1,:]  = P[h+1][w+2] - 2P[h+1][w+1] + P[h+1][w]
//  q=1: dy2[:,:,1:-1]  = P[h+2][w+1] - 2P[h+1][w+1] + P[h][w+1]
//  q=2,3: dxdy==dydx   = P[h+1][w+1] - P[h+1][w] - P[h][w+1] + P[h][w]
// ---------------------------------------------------------------------------
__global__ __launch_bounds__(256) void k_dd(const float* __restrict__ P,
                                            float* __restrict__ dd) {
  long long idx = (long long)blockIdx.x * 256 + threadIdx.x;
  const long long total = (long long)NIMG * HV * HV;
  if (idx >= total) return;
  int w = (int)(idx % HV);
  long long t = idx / HV;
  int h = (int)(t % HV);
  int im = (int)(t / HV);
  int c12 = im % 12, n = im / 12;
  int q = c12 / 3, c = c12 % 3;
  const float* p = P + (size_t)(n * 3 + c) * IMG_S;
  float v;
  if (q == 0)
    v = p[(h + 1) * IMG + (w + 2)] - 2.f * p[(h + 1) * IMG + (w + 1)] +
        p[(h + 1) * IMG + w];
  else if (q == 1)
    v = p[(h + 2) * IMG + (w + 1)] - 2.f * p[(h + 1) * IMG + (w + 1)] +
        p[h * IMG + (w + 1)];
  else
    v = p[(h + 1) * IMG + (w + 1)] - p[(h + 1) * IMG + w] -
        p[h * IMG + (w + 1)] + p[h * IMG + w];
  dd[(size_t)im * IMG_S + (size_t)h * IMG + w] = v;
}

// ---------------------------------------------------------------------------
// K0: per-channel min / max (for quantization levels).
// ---------------------------------------------------------------------------
__global__ __launch_bounds__(256) void k_minmax(const float* __restrict__ dd,
                                                float* __restrict__ mm) {
  __shared__ float smin[256], smax[256];
  const int img = blockIdx.x;
  const float* d = dd + (size_t)img * IMG_S;
  float lo = 3.4e38f, hi = -3.4e38f;
  for (int e = threadIdx.x; e < HV * HV; e += 256) {
    float v = d[(e / HV) * IMG + (e % HV)];
    lo = fminf(lo, v);
    hi = fmaxf(hi, v);
  }
  smin[threadIdx.x] = lo;
  smax[threadIdx.x] = hi;
  __syncthreads();
  for (int s = 128; s > 0; s >>= 1) {
    if (threadIdx.x < s) {
      smin[threadIdx.x] = fminf(smin[threadIdx.x], smin[threadIdx.x + s]);
      smax[threadIdx.x] = fmaxf(smax[threadIdx.x], smax[threadIdx.x + s]);
    }
    __syncthreads();
  }
  if (threadIdx.x == 0) {
    mm[img * 2 + 0] = smin[0];
    mm[img * 2 + 1] = smax[0];
  }
}

// ---------------------------------------------------------------------------
// K3: horizontal pass.  T_q = F_q(dd) * M   (F_q in {Wt, Wt*J, Wt*J^2}).
// Raw 16x512 dd strip is DMA'd into LDS by the TDM (2-dword pads every 256
// dwords -> row stride 516), then transformed into a second LDS buffer with
// odd stride 521 (adjacent + conflict-free WMMA A reads).  One wave per
// 16x16 output tile; B operand = banded mask from lane indices.  Writes the
// full padded 512x512 (zeros outside valid 510x510) so the vertical pass
// needs no guards.
// grid = (32 row-tiles, 96 images, 3 quantities), block = 256 (8 waves)
// ---------------------------------------------------------------------------
__global__ __launch_bounds__(256) void k_hpass(const float* __restrict__ dd,
                                               const float* __restrict__ mm,
                                               float* __restrict__ t0,
                                               float* __restrict__ t1,
                                               float* __restrict__ t2,
                                               int bin) {
  __shared__ float lds_raw[16 * RAWP];
  __shared__ float lds_f[16 * FP];
  const int img = blockIdx.y, q = blockIdx.z, h0 = blockIdx.x * 16;
  const float minc = mm[img * 2], maxc = mm[img * 2 + 1];
  const float scale = maxc - minc + 1e-5f;
  const float qv = minc + scale * ((float)bin * 0.1f);
  const float* dimg = dd + (size_t)img * IMG_S;

  // --- TDM: async DMA of the raw 16x512 strip into LDS (wave 0 issues) ---
  if (threadIdx.x < 32) {
    unsigned lds_off = (unsigned)(uintptr_t)(void*)lds_raw;  // flat[31:0]
    unsigned long long ga =
        (unsigned long long)(uintptr_t)(dimg + (size_t)h0 * IMG);
    v4u g0;
    g0[0] = 1u;  // count=1 (valid user descriptor), no gather
    g0[1] = lds_off;
    g0[2] = (unsigned)(ga & 0xffffffffu);
    g0[3] = (unsigned)((ga >> 32) & 0x01ffffffu) | 0x80000000u;  // type=2
    v8i g1;
    // data_size=4B, pad_enable, pad_interval=256 dwords, pad_amount=2 dwords
    g1[0] = (int)((2u << 16) | (1u << 20) | (7u << 22) | (1u << 25));
    g1[1] = (int)(512u << 16);  // tensor_dim0 = 512 (bits 79:48, low half)
    g1[2] = (int)(16u << 16);   // tensor_dim1 = 16   (bits 111:80, low half)
    g1[3] = (int)(512u << 16);  // tile_dim0  = 512   (bits 127:112)
    g1[4] = 16;                 // tile_dim1  = 16    (bits 143:128)
    g1[5] = 512;                // tensor_dim0_stride = 512 (bits 207:160)
    g1[6] = 0;
    g1[7] = 0;
    v4i z4 = {0, 0, 0, 0};
#if defined(__clang_major__) && (__clang_major__ >= 23)
    v8i z8 = {0, 0, 0, 0, 0, 0, 0, 0};
    __builtin_amdgcn_tensor_load_to_lds(g0, g1, z4, z4, z8, 0);
#else
    __builtin_amdgcn_tensor_load_to_lds(g0, g1, z4, z4, 0);
#endif
    __builtin_amdgcn_s_wait_tensorcnt(0);
  }
  __syncthreads();

  // --- transform raw J -> F_q into the clean-stride buffer (zero padding) ---
  for (int idx = threadIdx.x; idx < 16 * FP; idx += 256) {
    int r = idx / FP, c = idx - r * FP;
    float v = 0.f;
    if (c < HV && (h0 + r) < HV) {
      float J = lds_raw[r * RAWP + c + (c >= 256 ? 2 : 0)];
      float Wt = powm12(fabsf(J - qv) + 1e-4f);
      v = (q == 0) ? Wt : (q == 1) ? Wt * J : Wt * J * J;
    }
    lds_f[idx] = v;
  }
  __syncthreads();

  const int lane = threadIdx.x & 31, wv = threadIdx.x >> 5;
  const int khalf = (lane >> 4) << 1;  // K offset 0 (lanes 0-15) or 2 (16-31)
  const int ln = lane & 15;
  float* out = ((q == 0) ? t0 : (q == 1) ? t1 : t2) + (size_t)img * IMG_S;

  for (int tile = wv; tile < 32; tile += 8) {
    const int w0 = tile * 16;
    // Wave-uniform loop bounds -> scalar (s_cbranch) loop around the WMMAs.
    const int klo = __builtin_amdgcn_readfirstlane(max(0, w0 - R_WIN));
    const int khi = __builtin_amdgcn_readfirstlane(min(512, w0 + 16 + R_WIN));
    const int ncol = w0 + ln;  // B column (output column)
    v8f acc = {0, 0, 0, 0, 0, 0, 0, 0};
    for (int kc = klo; kc < khi; kc += 4) {
      int k0 = kc + khalf;
      v2f a, b;
      a.x = lds_f[ln * FP + k0];      // A(m=ln, k=k0)
      a.y = lds_f[ln * FP + k0 + 1];  // A(m=ln, k=k0+1)
      b.x = (iabs(k0 - ncol) <= R_WIN) ? 1.f : 0.f;      // M[k0, ncol]
      b.y = (iabs(k0 + 1 - ncol) <= R_WIN) ? 1.f : 0.f;  // M[k0+1, ncol]
      acc = wmma_f32x4(a, b, acc);
    }
    const int hbase = h0 + ((lane >> 4) << 3);  // D rows: r / r+8
    const int ww = w0 + ln;
#pragma unroll
    for (int r = 0; r < 8; ++r) {
      int hh = hbase + r;
      // Unguarded store over the padded image; zeros outside valid region.
      out[hh * IMG + ww] = (hh < HV && ww < HV) ? acc[r] : 0.f;
    }
  }
}

// ---------------------------------------------------------------------------
// K4: vertical pass + epilogue.  S_q = M * T_q ; then
//   loss = S0*J^2 - 2*J*S1 + S2, weighted by the bin interpolation factor
//   and accumulated into loss_acc / W_acc.  T_q padding is exact zeros, so
//   every load and store is unguarded (pad outputs written, never read).
// grid = (4 col-groups, 32 row-tiles, 96 images), block = 256 (8 waves),
// each wave owns one 16x16 output tile. A operand = generated mask.
// ---------------------------------------------------------------------------
__global__ __launch_bounds__(256) void k_vpass(const float* __restrict__ dd,
                                               const float* __restrict__ mm,
                                               const float* __restrict__ t0,
                                               const float* __restrict__ t1,
                                               const float* __restrict__ t2,
                                               float* __restrict__ lacc,
                                               float* __restrict__ wacc,
                                               int bin, int beta) {
  const int img = blockIdx.z, h0 = blockIdx.y * 16;
  const int lane = threadIdx.x & 31, wv = threadIdx.x >> 5;
  const int w0 = (blockIdx.x * 8 + wv) * 16;
  const int khalf = (lane >> 4) << 1;
  const int ln = lane & 15;
  const size_t ioff = (size_t)img * IMG_S;
  const float* p0 = t0 + ioff;
  const float* p1 = t1 + ioff;
  const float* p2 = t2 + ioff;

  // Wave-uniform (block-uniform, in fact) scalar loop bounds.
  const int klo = __builtin_amdgcn_readfirstlane(max(0, h0 - R_WIN));
  const int khi = __builtin_amdgcn_readfirstlane(min(512, h0 + 16 + R_WIN));
  const int mg = h0 + ln;   // A row = output row
  const int col = w0 + ln;  // B column

  v8f aW = {0, 0, 0, 0, 0, 0, 0, 0};
  v8f aI = {0, 0, 0, 0, 0, 0, 0, 0};
  v8f aI2 = {0, 0, 0, 0, 0, 0, 0, 0};

  for (int kc = klo; kc < khi; kc += 4) {
    const int k0 = kc + khalf, k1 = k0 + 1;
    v2f a, b;
    a.x = (iabs(mg - k0) <= R_WIN) ? 1.f : 0.f;  // M[mg, k0]
    a.y = (iabs(mg - k1) <= R_WIN) ? 1.f : 0.f;  // M[mg, k1]
    // prefetch next chunk's row (lowers to global_prefetch_b8)
    __builtin_prefetch(p0 + (size_t)min(kc + 4, 511) * IMG + col, 0, 0);
    b.x = p0[k0 * IMG + col];
    b.y = p0[k1 * IMG + col];
    aW = wmma_f32x4(a, b, aW);
    b.x = p1[k0 * IMG + col];
    b.y = p1[k1 * IMG + col];
    aI = wmma_f32x4(a, b, aI);
    b.x = p2[k0 * IMG + col];
    b.y = p2[k1 * IMG + col];
    aI2 = wmma_f32x4(a, b, aI2);
  }

  const float minc = mm[img * 2], maxc = mm[img * 2 + 1];
  const float scale = maxc - minc + 1e-5f;
  const float* dimg = dd + ioff;
  const int hbase = h0 + ((lane >> 4) << 3);
#pragma unroll
  for (int r = 0; r < 8; ++r) {
    const int hh = hbase + r;
    const size_t o = ioff + (size_t)hh * IMG + col;
    float J = dimg[hh * IMG + col];  // pad region: finite garbage, never read
    float tt = (J - minc) / scale * 10.f;
    int b1 = (int)floorf(tt), b2 = (int)ceilf(tt);
    float wgt = 0.f;
    if (bin < NBINS && b1 == bin) {  // acts as loss1 in iteration bin+1
      float Qa = minc + scale * ((float)bin * 0.1f);
      float Qb = minc + scale * ((float)(bin + 1) * 0.1f);
      wgt += (Qb - J) / (Qb - Qa);
    }
    if (bin >= 1 && b2 == bin) {  // acts as loss2 in iteration bin
      float Qa = minc + scale * ((float)(bin - 1) * 0.1f);
      float Qb = minc + scale * ((float)bin * 0.1f);
      wgt += (J - Qa) / (Qb - Qa);
    }
    float W = aW[r];
    float lossv = W * J * J - 2.f * J * aI[r] + aI2[r];
    if (beta) {
      lacc[o] += wgt * lossv;
      wacc[o] += wgt * W;
    } else {
      lacc[o] = wgt * lossv;
      wacc[o] = wgt * W;
    }
  }
}

// ---------------------------------------------------------------------------
// K5: deterministic two-stage reduction of mean(loss_acc / W_acc) per batch.
// ---------------------------------------------------------------------------
__global__ __launch_bounds__(256) void k_red1(const float* __restrict__ lacc,
                                              const float* __restrict__ wacc,
                                              float* __restrict__ part) {
  __shared__ float s[256];
  const long long per_n = 12LL * HV * HV;
  const int n = blockIdx.y;
  float acc = 0.f;
  for (long long e = (long long)blockIdx.x * 256 + threadIdx.x; e < per_n;
       e += 128LL * 256) {
    int c = (int)(e / (HV * HV));
    int rem = (int)(e % (HV * HV));
    int h = rem / HV, w = rem % HV;
    size_t o = (size_t)(n * 12 + c) * IMG_S + (size_t)h * IMG + w;
    acc += lacc[o] / wacc[o];
  }
  s[threadIdx.x] = acc;
  __syncthreads();
  for (int st = 128; st > 0; st >>= 1) {
    if (threadIdx.x < st) s[threadIdx.x] += s[threadIdx.x + st];
    __syncthreads();
  }
  if (threadIdx.x == 0) part[n * 128 + blockIdx.x] = s[0];
}

__global__ __launch_bounds__(128) void k_red2(const float* __restrict__ part,
                                              float* __restrict__ out) {
  __shared__ float s[128];
  const int n = blockIdx.x;
  s[threadIdx.x] = part[n * 128 + threadIdx.x];
  __syncthreads();
  for (int st = 64; st > 0; st >>= 1) {
    if (threadIdx.x < st) s[threadIdx.x] += s[threadIdx.x + st];
    __syncthreads();
  }
  if (threadIdx.x == 0) out[n] = s[0] / (float)(12LL * HV * HV);
}

// ---------------------------------------------------------------------------
extern "C" void kernel_launch(void* const* d_in, const int* in_sizes, int n_in,
                              void* d_out, int out_size, void* d_ws,
                              size_t ws_size, hipStream_t stream) {
  (void)in_sizes; (void)n_in; (void)out_size; (void)ws_size;
  const float* P = (const float*)d_in[0];
  float* out = (float*)d_out;
  float* ws = (float*)d_ws;

  const size_t S = (size_t)NIMG * IMG_S;  // floats per full buffer
  float* mm = ws;                 // 192 floats (min/max per channel)
  float* part = ws + 256;         // 8 * 128 partial sums
  float* dd = ws + 2048;          // 96 MiB
  float* t0 = dd + S;             // 96 MiB each
  float* t1 = t0 + S;
  float* t2 = t1 + S;
  float* lacc = t2 + S;
  float* wacc = lacc + S;

  {
    const long long total = (long long)NIMG * HV * HV;
    const int nb = (int)((total + 255) / 256);
    k_dd<<<nb, 256, 0, stream>>>(P, dd);
  }
  k_minmax<<<NIMG, 256, 0, stream>>>(dd, mm);

  for (int bin = 0; bin <= NBINS; ++bin) {
    k_hpass<<<dim3(32, NIMG, 3), 256, 0, stream>>>(dd, mm, t0, t1, t2, bin);
    k_vpass<<<dim3(4, 32, NIMG), 256, 0, stream>>>(dd, mm, t0, t1, t2, lacc,
                                                   wacc, bin, bin > 0 ? 1 : 0);
  }

  k_red1<<<dim3(128, 8), 256, 0, stream>>>(lacc, wacc, part);
  k_red2<<<8, 128, 0, stream>>>(part, out);
}